// SingleDimHistLayer_33895881900581
// MI455X (gfx1250) — compile-verified
//
#include <hip/hip_runtime.h>
#include <hip/hip_bf16.h>

// Soft histogram (sigmoid kernel), B=8, N=512*512, K=256.
//   t = 640*(x - mu_k);  P = sigmoid(t+1.25) - sigmoid(t-1.25)
// Closed form with a = e^1.25:  P = e*(a-1/a) / (1 + e*(a+1/a) + e^2), e = exp(-t)
// exp(-t) = exp2((mu_k - x)*640*log2e) = exp2((k+0.5)*3.6067376 - x*923.3248262)
//
// Kernel 1: band-limited (18-bin window) scatter into per-wave LDS histograms,
//           per-block 256-bin partials written to d_ws.
// Kernel 2: cross-block reduction as GEMM on the matrix pipe:
//           V_WMMA_F32_16X16X4_F32 with B = ones (full f32 precision).

typedef __attribute__((ext_vector_type(2))) float v2f;
typedef __attribute__((ext_vector_type(8))) float v8f;

#define KBINS       256
#define NPIX        262144        // 512*512 per batch
#define BPB         32            // blocks per batch
#define PX_PER_BLK  (NPIX / BPB)  // 8192
#define WAVES_PER_BLK 8

__global__ __launch_bounds__(256) void hist_partial_kernel(
    const float* __restrict__ x, float* __restrict__ partials) {
  __shared__ float hist[WAVES_PER_BLK][KBINS];

  const int tid  = threadIdx.x;          // 0..255
  const int wave = tid >> 5;             // 0..7 (wave32)

  // zero the replicated LDS histograms
  #pragma unroll
  for (int i = 0; i < WAVES_PER_BLK; ++i) hist[i][tid] = 0.0f;
  __syncthreads();

  const int blk = blockIdx.x;                       // batch*BPB + chunk
  const size_t base = (size_t)blk * PX_PER_BLK;     // 32*8192 == 262144 per batch
  const float4* xv = (const float4*)(x + base);
  float* h = hist[wave];

  const float C_E   = 923.3248262f;   // 640*log2(e)
  const float C_MU  = 3.6067376f;     // C_E / 256
  const float C1    = 3.7768477543f;  // e^1.25 + e^-1.25
  const float C2    = 3.2038381606f;  // e^1.25 - e^-1.25

  for (int it = 0; it < PX_PER_BLK / (4 * 256); ++it) {   // 8 iterations
    float4 px = xv[it * 256 + tid];
    float vals[4] = {px.x, px.y, px.z, px.w};
    #pragma unroll
    for (int q = 0; q < 4; ++q) {
      const float xx = vals[q];
      const float xs = xx * C_E;
      const int   kc = (int)floorf(xx * 256.0f - 0.5f);  // nearest-bin coordinate
      const float zb = ((float)kc - 7.5f) * C_MU - xs;   // z at k = kc-8
      #pragma unroll
      for (int j = 0; j < 18; ++j) {
        float z = zb + (float)j * C_MU;                  // (k+0.5)*C_MU - x*C_E
        z = __builtin_amdgcn_fmed3f(z, -40.0f, 40.0f);   // keep e, e^2 finite
        const float e   = __builtin_amdgcn_exp2f(z);
        const float den = fmaf(e, e, fmaf(e, C1, 1.0f));
        const float p   = (e * C2) * __builtin_amdgcn_rcpf(den);
        const int k = kc - 8 + j;
        if (k >= 0 && k < KBINS) atomicAdd(&h[k], p);    // ds_add_f32 (no return)
      }
    }
  }
  __syncthreads();

  // fold 8 wave-replicas, emit this block's 256-bin partial
  float s = 0.0f;
  #pragma unroll
  for (int w = 0; w < WAVES_PER_BLK; ++w) s += hist[w][tid];
  partials[(size_t)blk * KBINS + tid] = s;
}

// Reduction-as-GEMM: out[b,k] = (1/N) * sum_{w=0..31} partials[b,w,k]
// One wave per (batch, 16-bin tile). A = 16x4 f32 (bins x partial-chunks),
// B = ones(4x16), D = f32 16x16 accumulator; every column of D holds the sums.
__global__ __launch_bounds__(512) void hist_reduce_wmma_kernel(
    const float* __restrict__ partials, float* __restrict__ out) {
  const int lane  = threadIdx.x & 31;
  const int tile  = threadIdx.x >> 5;   // 0..15  (16 waves/block)
  const int batch = blockIdx.x;         // 0..7
  const int m     = lane & 15;          // row M of the A-matrix this lane feeds
  const int bin   = tile * 16 + m;
  // ISA 32-bit A 16x4 layout: lanes 0-15 carry K=0,1 ; lanes 16-31 carry K=2,3
  const int wsel  = (lane < 16) ? 0 : 2;

  const float* base = partials + ((size_t)batch * BPB) * KBINS + bin;

  v8f acc = {0.f, 0.f, 0.f, 0.f, 0.f, 0.f, 0.f, 0.f};
  v2f bones; bones.x = 1.0f; bones.y = 1.0f;   // B = all-ones 4x16

  #pragma unroll
  for (int s = 0; s < BPB / 4; ++s) {          // 8 WMMAs: 4 partials each
    const int w0 = s * 4 + wsel;
    v2f a;
    a.x = base[(size_t)w0 * KBINS];
    a.y = base[(size_t)(w0 + 1) * KBINS];
    acc = __builtin_amdgcn_wmma_f32_16x16x4_f32(
        /*neg_a=*/false, a, /*neg_b=*/false, bones,
        /*c_mod=*/(short)0, acc, /*reuse_a=*/false, /*reuse_b=*/false);
  }

  // D layout: lane 0 holds M=0..7 (VGPR 0..7) at N=0; lane 16 holds M=8..15.
  if (m == 0) {
    const int   mbase = (lane == 0) ? 0 : 8;
    const float invN  = 1.0f / (float)NPIX;
    #pragma unroll
    for (int i = 0; i < 8; ++i)
      out[batch * KBINS + tile * 16 + mbase + i] = acc[i] * invN;
  }
}

extern "C" void kernel_launch(void* const* d_in, const int* in_sizes, int n_in,
                              void* d_out, int out_size, void* d_ws, size_t ws_size,
                              hipStream_t stream) {
  const float* x   = (const float*)d_in[0];   // [8,512,512] f32
  float*       out = (float*)d_out;           // [8,256] f32
  float*       ws  = (float*)d_ws;            // needs 8*32*256*4 = 256 KiB
  (void)in_sizes; (void)n_in; (void)out_size; (void)ws_size;

  hist_partial_kernel<<<8 * BPB, 256, 0, stream>>>(x, ws);
  hist_reduce_wmma_kernel<<<8, 512, 0, stream>>>(ws, out);
}